// PermOnlySSMLayer_62302795596528
// MI455X (gfx1250) — compile-verified
//
#include <hip/hip_runtime.h>
#include <hip/hip_bf16.h>

typedef __attribute__((ext_vector_type(2))) float v2f;
typedef __attribute__((ext_vector_type(8))) float v8f;

#define Bsz 4
#define Ssz 1024
#define Dsz 1024
#define Hsz 8
#define Nst 16

// ---------------------------------------------------------------------------
// LayerNorm: one 256-thread block per row of 1024 elements.
// ---------------------------------------------------------------------------
__global__ __launch_bounds__(256)
void layernorm_kernel(const float* __restrict__ x,
                      const float* __restrict__ w,
                      const float* __restrict__ b,
                      float* __restrict__ xn) {
  __shared__ float red[256];
  const int tid = threadIdx.x;
  const size_t row = blockIdx.x;
  const float4* xr = (const float4*)(x + row * Dsz);
  float4 v = xr[tid];
  float s  = v.x + v.y + v.z + v.w;
  float sq = v.x * v.x + v.y * v.y + v.z * v.z + v.w * v.w;

  red[tid] = s; __syncthreads();
  for (int off = 128; off > 0; off >>= 1) {
    if (tid < off) red[tid] += red[tid + off];
    __syncthreads();
  }
  float mean = red[0] * (1.0f / Dsz);
  __syncthreads();
  red[tid] = sq; __syncthreads();
  for (int off = 128; off > 0; off >>= 1) {
    if (tid < off) red[tid] += red[tid + off];
    __syncthreads();
  }
  float var = red[0] * (1.0f / Dsz) - mean * mean;
  float inv = rsqrtf(var + 1e-5f);

  float4 wv = ((const float4*)w)[tid];
  float4 bv = ((const float4*)b)[tid];
  float4 o;
  o.x = (v.x - mean) * inv * wv.x + bv.x;
  o.y = (v.y - mean) * inv * wv.y + bv.y;
  o.z = (v.z - mean) * inv * wv.z + bv.z;
  o.w = (v.w - mean) * inv * wv.w + bv.w;
  ((float4*)(xn + row * Dsz))[tid] = o;
}

// ---------------------------------------------------------------------------
// f32 WMMA GEMM.  out[M,Nn] = A[M,K] @ W[K,Nn] + bias (+ residual)
// Block = 8 waves = 128 rows x 64 cols macro-tile, double-buffered k-blocks:
//   A: ASYNC global->LDS b128 (k-contiguous, so fragments stay adjacent).
//      As[rowLocal][k], row stride 36 dwords: 16B-aligned stores, banks
//      36r mod 64 distinct, kb=2 half-wave on a disjoint bank set.
//   B: global->VGPR->transposed LDS Bs[col][k], stride 34 dwords -> each
//      fragment {W[k][c],W[k+1][c]} is one aligned b64 LDS read, no movs.
// Inner loop is pure LDS: per kk -> 1 A b64 + CT B b64 + CT WMMA.
// A fragment (16x4 f32): lane l -> row=l&15, kb=(l>>4)*2, holds K=kb,kb+1.
// B fragment (4x16 f32): lane l -> col=l&15, kb=(l>>4)*2, holds K=kb,kb+1.
// C/D: vgpr v, lane l -> row = v + 8*(l>>4), col = l&15.
// ---------------------------------------------------------------------------
template <int CT>
__global__ __launch_bounds__(256)
void gemm_f32_wmma(const float* __restrict__ A,
                   const float* __restrict__ W,
                   const float* __restrict__ bias,
                   const float* __restrict__ residual,
                   float* __restrict__ out,
                   int M, int K, int Nn) {
  constexpr int KC  = 32;   // k-block depth
  constexpr int LDB = 34;   // B tile: [col][k] padded
  constexpr int LDA = 36;   // A tile: [rowLocal][k] padded (mult. of 4)
  __shared__ float Bs[2][16 * CT * LDB];
  __shared__ float As[2][128 * LDA];

  const int colStrips = Nn / (16 * CT);
  const int rowBlock  = blockIdx.x / colStrips;
  const int colStrip  = blockIdx.x % colStrips;
  const int col0 = colStrip * 16 * CT;

  const int waveId = threadIdx.x >> 5;
  const int lane   = threadIdx.x & 31;

  const int r  = lane & 15;
  const int kb = (lane >> 4) * 2;

  // B staging split: thread t -> cols (t&15)*4..+3 of k rows (t>>4), (t>>4)+16
  const int sc = (threadIdx.x & 15) * 4;
  const int sk = threadIdx.x >> 4;      // 0..15
  // A staging split: thread t -> row t>>1, k quads (t&1)*4 + {0,8,16,24}
  const int arow = threadIdx.x >> 1;    // 0..127
  const int ak   = (threadIdx.x & 1) * 4;

  v8f acc[CT];
#pragma unroll
  for (int t = 0; t < CT; ++t)
#pragma unroll
    for (int v = 0; v < 8; ++v) acc[t][v] = 0.0f;

  const int nblk = K / KC;

  const uint32_t aLds0      = (uint32_t)(uintptr_t)&As[0][arow * LDA + ak];
  const uint32_t aBufStride = (uint32_t)(128 * LDA * sizeof(float));
  const uint32_t aRowGlob   = (uint32_t)((rowBlock * 128 + arow) * K);  // elems

  // ---- prologue: stage k-block 0 into buffer 0 ----
  {
#pragma unroll
    for (int q = 0; q < 4; ++q) {
      uint32_t goff = (aRowGlob + (uint32_t)(ak + q * 8)) * (uint32_t)sizeof(float);
      asm volatile("global_load_async_to_lds_b128 %0, %1, %2"
                   :: "v"(aLds0 + (uint32_t)(q * 8 * sizeof(float))),
                      "v"(goff), "s"(A) : "memory");
    }
    float4 s0 = *(const float4*)&W[(size_t)sk * Nn + col0 + sc];
    float4 s1 = *(const float4*)&W[(size_t)(sk + 16) * Nn + col0 + sc];
#pragma unroll
    for (int j = 0; j < 4; ++j) {
      Bs[0][(sc + j) * LDB + sk]      = ((const float*)&s0)[j];
      Bs[0][(sc + j) * LDB + sk + 16] = ((const float*)&s1)[j];
    }
  }
  asm volatile("s_wait_asynccnt 0x0" ::: "memory");
  __syncthreads();

  int buf = 0;
  for (int blk = 0; blk < nblk; ++blk) {
    const int kb0 = blk * KC;
    const bool more = (blk + 1 < nblk);

    // stage NEXT k-block: A via async copies, B into registers
    float4 n0, n1;
    if (more) {
      const uint32_t nb = (uint32_t)(buf ^ 1);
#pragma unroll
      for (int q = 0; q < 4; ++q) {
        uint32_t goff = (aRowGlob + (uint32_t)(kb0 + KC + ak + q * 8)) *
                        (uint32_t)sizeof(float);
        asm volatile("global_load_async_to_lds_b128 %0, %1, %2"
                     :: "v"(aLds0 + nb * aBufStride + (uint32_t)(q * 8 * sizeof(float))),
                        "v"(goff), "s"(A) : "memory");
      }
      n0 = *(const float4*)&W[(size_t)(kb0 + KC + sk) * Nn + col0 + sc];
      n1 = *(const float4*)&W[(size_t)(kb0 + KC + sk + 16) * Nn + col0 + sc];
    }

    // compute this k-block entirely from LDS
    const float* asb = &As[buf][0];
    const float* bsb = &Bs[buf][0];
    const int abase = (waveId * 16 + r) * LDA + kb;
#pragma unroll
    for (int kk = 0; kk < KC / 4; ++kk) {
      v2f af = *(const v2f*)&asb[abase + kk * 4];
#pragma unroll
      for (int t = 0; t < CT; ++t) {
        v2f bf = *(const v2f*)&bsb[(t * 16 + r) * LDB + kk * 4 + kb];
        acc[t] = __builtin_amdgcn_wmma_f32_16x16x4_f32(
            /*neg_a=*/false, af, /*neg_b=*/false, bf,
            /*c_mod=*/(short)0, acc[t], /*reuse_a=*/false, /*reuse_b=*/false);
      }
    }

    if (more) {
#pragma unroll
      for (int j = 0; j < 4; ++j) {
        Bs[buf ^ 1][(sc + j) * LDB + sk]      = ((const float*)&n0)[j];
        Bs[buf ^ 1][(sc + j) * LDB + sk + 16] = ((const float*)&n1)[j];
      }
      asm volatile("s_wait_asynccnt 0x0" ::: "memory");
      __syncthreads();
      buf ^= 1;
    }
  }

  // epilogue: bias (+residual) + store
  const int rbase = rowBlock * 128 + waveId * 16 + 8 * (lane >> 4);
#pragma unroll
  for (int t = 0; t < CT; ++t) {
    const int ocol = col0 + t * 16 + r;
    const float bv = bias ? bias[ocol] : 0.0f;
#pragma unroll
    for (int v = 0; v < 8; ++v) {
      const size_t orow = (size_t)(rbase + v);
      float val = acc[t][v] + bv;
      if (residual) val += residual[orow * Nn + ocol];
      out[orow * Nn + ocol] = val;
    }
  }
}

// ---------------------------------------------------------------------------
// Sinkhorn (in place): 5 iterations of row-LSE then column-LSE subtraction,
// then exp. 16 threads per 16x16 matrix; 16 matrices per 256-thread block.
// ---------------------------------------------------------------------------
__global__ __launch_bounds__(256)
void sinkhorn_kernel(float* __restrict__ soft) {
  __shared__ float tile[16][16][17];
  __shared__ float collse[16][16];
  const int mloc = threadIdx.x >> 4;
  const int r    = threadIdx.x & 15;
  const size_t mm = (size_t)blockIdx.x * 16 + mloc;
  float* base = soft + mm * 256 + r * 16;

  float rv[16];
  float4* b4 = (float4*)base;
#pragma unroll
  for (int q = 0; q < 4; ++q) {
    float4 t = b4[q];
    rv[q * 4 + 0] = t.x; rv[q * 4 + 1] = t.y;
    rv[q * 4 + 2] = t.z; rv[q * 4 + 3] = t.w;
  }

  for (int it = 0; it < 5; ++it) {
    // row logsumexp
    float mx = rv[0];
#pragma unroll
    for (int j = 1; j < 16; ++j) mx = fmaxf(mx, rv[j]);
    float sum = 0.0f;
#pragma unroll
    for (int j = 0; j < 16; ++j) sum += __expf(rv[j] - mx);
    float lse = mx + __logf(sum);
#pragma unroll
    for (int j = 0; j < 16; ++j) {
      rv[j] -= lse;
      tile[mloc][r][j] = rv[j];
    }
    __syncthreads();
    // column logsumexp: thread r handles column r of its matrix
    float cmx = tile[mloc][0][r];
#pragma unroll
    for (int i = 1; i < 16; ++i) cmx = fmaxf(cmx, tile[mloc][i][r]);
    float cs = 0.0f;
#pragma unroll
    for (int i = 0; i < 16; ++i) cs += __expf(tile[mloc][i][r] - cmx);
    collse[mloc][r] = cmx + __logf(cs);
    __syncthreads();
#pragma unroll
    for (int j = 0; j < 16; ++j) rv[j] -= collse[mloc][j];
  }

#pragma unroll
  for (int q = 0; q < 4; ++q) {
    float4 t;
    t.x = __expf(rv[q * 4 + 0]); t.y = __expf(rv[q * 4 + 1]);
    t.z = __expf(rv[q * 4 + 2]); t.w = __expf(rv[q * 4 + 3]);
    b4[q] = t;
  }
}

// ---------------------------------------------------------------------------
// Sequential scan over S: one wave per (b,h). Lane i owns state element i.
// h_new = sigmoid(g) * (P h) + (1 - sigmoid(g)) * b_in.
// Next-step operands are prefetched while the current dot product executes.
// ---------------------------------------------------------------------------
__global__ __launch_bounds__(32)
void scan_kernel(const float* __restrict__ soft,
                 const float* __restrict__ gpre,
                 const float* __restrict__ bin,
                 float* __restrict__ hs) {
  const int bh = blockIdx.x;        // 0..31
  const int b  = bh >> 3;
  const int h  = bh & 7;
  const int lane = threadIdx.x;     // 0..31
  const int i = lane & 15;

  const float4* prow = (const float4*)(soft + (((size_t)b * Ssz) * Hsz + h) * 256 + i * 16);
  const float*  gp   = gpre + (size_t)b * Ssz * (Hsz * Nst) + h * Nst + i;
  const float*  bp   = bin  + (size_t)b * Ssz * (Hsz * Nst) + h * Nst + i;

  float hv = 0.0f;
  float4 p0 = prow[0], p1 = prow[1], p2 = prow[2], p3 = prow[3];
  float gg = *gp, bb = *bp;

  for (int s = 0; s < Ssz; ++s) {
    float4 n0, n1, n2, n3;
    float ng = 0.0f, nb = 0.0f;
    if (s + 1 < Ssz) {
      const float4* pn = prow + 512;  // next step: +2048 floats
      n0 = pn[0]; n1 = pn[1]; n2 = pn[2]; n3 = pn[3];
      ng = gp[Hsz * Nst];
      nb = bp[Hsz * Nst];
    } else {
      n0 = p0; n1 = p1; n2 = p2; n3 = p3;
    }

    float p[16] = {p0.x, p0.y, p0.z, p0.w, p1.x, p1.y, p1.z, p1.w,
                   p2.x, p2.y, p2.z, p2.w, p3.x, p3.y, p3.z, p3.w};
    float dot = 0.0f;
#pragma unroll
    for (int j = 0; j < 16; ++j) {
      float hj = __shfl(hv, j, 32);
      dot = fmaf(p[j], hj, dot);
    }
    float g = 1.0f / (1.0f + __expf(-gg));
    hv = g * dot + (1.0f - g) * bb;

    if (lane < 16)
      hs[(((size_t)b * Ssz + s) * Hsz + h) * Nst + i] = hv;

    prow += 512; gp += Hsz * Nst; bp += Hsz * Nst;
    p0 = n0; p1 = n1; p2 = n2; p3 = n3; gg = ng; bb = nb;
  }
}

// ---------------------------------------------------------------------------
// Launcher
// ---------------------------------------------------------------------------
extern "C" void kernel_launch(void* const* d_in, const int* in_sizes, int n_in,
                              void* d_out, int out_size, void* d_ws, size_t ws_size,
                              hipStream_t stream) {
  const float* x    = (const float*)d_in[0];
  const float* ln_w = (const float*)d_in[1];
  const float* ln_b = (const float*)d_in[2];
  const float* Wp   = (const float*)d_in[3];
  const float* bp   = (const float*)d_in[4];
  const float* Wg   = (const float*)d_in[5];
  const float* bg   = (const float*)d_in[6];
  const float* Wi   = (const float*)d_in[7];
  const float* bi   = (const float*)d_in[8];
  const float* Wo   = (const float*)d_in[9];
  const float* bo   = (const float*)d_in[10];
  float* out = (float*)d_out;

  const size_t M = (size_t)Bsz * Ssz;          // 4096 rows
  float* ws   = (float*)d_ws;
  float* xn   = ws;                             // [4096,1024]
  float* soft = xn   + M * Dsz;                 // [4096,2048]
  float* gpre = soft + M * (Hsz * Nst * Nst);   // [4096,128]
  float* binb = gpre + M * (Hsz * Nst);         // [4096,128]
  float* hsb  = binb + M * (Hsz * Nst);         // [4096,128]

  // 1) LayerNorm
  layernorm_kernel<<<(int)M, 256, 0, stream>>>(x, ln_w, ln_b, xn);

  // 2) Projections (WMMA f32; A async-LDS, B transposed-LDS, double-buffered)
  gemm_f32_wmma<4><<<(int)(M / 128) * (2048 / 64), 256, 0, stream>>>(
      xn, Wp, bp, nullptr, soft, (int)M, Dsz, 2048);
  gemm_f32_wmma<4><<<(int)(M / 128) * (128 / 64), 256, 0, stream>>>(
      xn, Wg, bg, nullptr, gpre, (int)M, Dsz, 128);
  gemm_f32_wmma<4><<<(int)(M / 128) * (128 / 64), 256, 0, stream>>>(
      xn, Wi, bi, nullptr, binb, (int)M, Dsz, 128);

  // 3) Sinkhorn over 32768 matrices, 16 per block
  sinkhorn_kernel<<<(int)(M * Hsz / 16), 256, 0, stream>>>(soft);

  // 4) Sequential scan: 32 (b,h) recurrences, one wave each
  scan_kernel<<<Bsz * Hsz, 32, 0, stream>>>(soft, gpre, binb, hsb);

  // 5) Output projection + bias + residual
  gemm_f32_wmma<4><<<(int)(M / 128) * (1024 / 64), 256, 0, stream>>>(
      hsb, Wo, bo, x, out, (int)M, 128, 1024);
}